// BevPoolV2_8478265442577
// MI455X (gfx1250) — compile-verified
//
#include <hip/hip_runtime.h>
#include <hip/hip_bf16.h>

// bev_pool_v2 on gfx1250: sorted-segment-sum expressed as banded sparse GEMM
// via V_WMMA_F32_16X16X4_F32. One wave32 = 16 bev cells x 16 channels.
// Main loop is peeled (full 4-point chunks only, no tail predicates) and
// 2x unrolled (8 points / 2 WMMAs per iteration) for gather MLP.

typedef float v2f __attribute__((ext_vector_type(2)));
typedef float v8f __attribute__((ext_vector_type(8)));

#define BEV_CELLS 16384   // B*DZ*DY*DX = 1*1*128*128
#define C_CH      80      // channels

__device__ __forceinline__ int lower_bound_i32(const int* __restrict__ a, int n, int key) {
    int lo = 0, hi = n;
    while (lo < hi) {
        int mid = (lo + hi) >> 1;
        if (a[mid] < key) lo = mid + 1; else hi = mid;
    }
    return lo;
}

__global__ __launch_bounds__(32)
void BevPoolV2_wmma_kernel(const float* __restrict__ depth,
                           const float* __restrict__ feat,
                           const int*   __restrict__ rd,
                           const int*   __restrict__ rf,
                           const int*   __restrict__ rb,
                           float*       __restrict__ out,
                           int n_points) {
    const int segbase = blockIdx.x * 16;       // 16 consecutive bev cells
    const int cb      = blockIdx.y * 16;       // 16-channel group
    const int lane    = threadIdx.x;           // wave32
    const int half    = lane >> 4;             // 0: lanes 0-15, 1: lanes 16-31
    const int l15     = lane & 15;

    // Point range covering bev values [segbase, segbase+16) in sorted rb.
    int p_lo = lower_bound_i32(rb, n_points, segbase);
    int p_hi = lower_bound_i32(rb, n_points, segbase + 16);
    // Force SGPR-uniform loop bounds so EXEC stays all-ones around WMMA.
    p_lo = __builtin_amdgcn_readfirstlane(p_lo);
    p_hi = __builtin_amdgcn_readfirstlane(p_hi);

    v8f acc = {0.f, 0.f, 0.f, 0.f, 0.f, 0.f, 0.f, 0.f};

    // A 16x4 layout: VGPR0 holds K = 2*half, VGPR1 holds K = 2*half+1; M = l15.
    // B 4x16 layout: VGPR0 holds K = 2*half, VGPR1 holds K = 2*half+1; N = l15.
    const int k0  = half * 2;
    const int myb = segbase + l15;             // bev value owned by this A-row
    const int cbl = cb + l15;                  // this lane's channel

    int p0 = p_lo;
    const int n_pts  = p_hi - p_lo;
    const int p_dual = p_lo + (n_pts & ~7);    // end of full 8-point chunks
    const int p_full = p_lo + (n_pts & ~3);    // end of full 4-point chunks

    // ---- 2x-unrolled main loop: 8 in-range points, 2 WMMAs, no tail preds.
    for (; p0 < p_dual; p0 += 8) {
        const int pa = p0 + k0;
        const int pb = pa + 1;
        const int pc = pa + 4;
        const int pd = pa + 5;

        const int ba = rb[pa], bb = rb[pb], bc = rb[pc], bd = rb[pd];
        const float da = depth[rd[pa]], db = depth[rd[pb]];
        const float dc = depth[rd[pc]], dd = depth[rd[pd]];

        const float b0 = feat[rf[pa] * C_CH + cbl];
        const float b1 = feat[rf[pb] * C_CH + cbl];
        const float b2 = feat[rf[pc] * C_CH + cbl];
        const float b3 = feat[rf[pd] * C_CH + cbl];

        v2f A0 = {(ba == myb) ? da : 0.0f, (bb == myb) ? db : 0.0f};
        v2f B0 = {b0, b1};
        v2f A1 = {(bc == myb) ? dc : 0.0f, (bd == myb) ? dd : 0.0f};
        v2f B1 = {b2, b3};

        acc = __builtin_amdgcn_wmma_f32_16x16x4_f32(
            false, A0, false, B0, (short)0, acc, false, false);
        acc = __builtin_amdgcn_wmma_f32_16x16x4_f32(
            false, A1, false, B1, (short)0, acc, false, false);
    }

    // ---- at most one remaining full 4-point chunk.
    if (p0 < p_full) {
        const int pa = p0 + k0;
        const int pb = pa + 1;
        const int ba = rb[pa], bb = rb[pb];
        const float da = depth[rd[pa]], db = depth[rd[pb]];
        const float b0 = feat[rf[pa] * C_CH + cbl];
        const float b1 = feat[rf[pb] * C_CH + cbl];
        v2f A = {(ba == myb) ? da : 0.0f, (bb == myb) ? db : 0.0f};
        v2f B = {b0, b1};
        acc = __builtin_amdgcn_wmma_f32_16x16x4_f32(
            false, A, false, B, (short)0, acc, false, false);
        p0 += 4;
    }

    // ---- masked tail chunk (1..3 live points). Clamped indices keep loads
    // in-bounds and B finite; A columns for p >= p_hi are forced to zero.
    if (p0 < p_hi) {
        const int pa  = p0 + k0;
        const int pb  = pa + 1;
        const int pac = (pa < p_hi) ? pa : (p_hi - 1);
        const int pbc = (pb < p_hi) ? pb : (p_hi - 1);
        const int   ba = rb[pac], bb = rb[pbc];
        const float da = depth[rd[pac]], db = depth[rd[pbc]];
        const float b0 = feat[rf[pac] * C_CH + cbl];
        const float b1 = feat[rf[pbc] * C_CH + cbl];
        v2f A = {((pa < p_hi) && (ba == myb)) ? da : 0.0f,
                 ((pb < p_hi) && (bb == myb)) ? db : 0.0f};
        v2f B = {b0, b1};
        acc = __builtin_amdgcn_wmma_f32_16x16x4_f32(
            false, A, false, B, (short)0, acc, false, false);
    }

    // C/D layout: VGPR j -> M = j + 8*half (bev row), N = l15 (channel).
    // Output is channel-major: out[c * BEV_CELLS + bev]. Rows j are
    // consecutive bev cells for a fixed lane -> two aligned b128 stores.
    float* o = out + (size_t)cbl * BEV_CELLS + segbase + half * 8;
    float4 lo4 = make_float4(acc[0], acc[1], acc[2], acc[3]);
    float4 hi4 = make_float4(acc[4], acc[5], acc[6], acc[7]);
    *(float4*)(o + 0) = lo4;
    *(float4*)(o + 4) = hi4;
}

extern "C" void kernel_launch(void* const* d_in, const int* in_sizes, int n_in,
                              void* d_out, int out_size, void* d_ws, size_t ws_size,
                              hipStream_t stream) {
    // setup_inputs order:
    // 0: depth (f32), 1: feat (f32), 2: ranks_depths (i32), 3: ranks_feats (i32),
    // 4: ranks_bevs (i32, sorted), 5: bev_feat_shape, 6: interval_starts,
    // 7: interval_lengths (intervals are redundant; we binary-search rb).
    const float* depth = (const float*)d_in[0];
    const float* feat  = (const float*)d_in[1];
    const int*   rd    = (const int*)d_in[2];
    const int*   rf    = (const int*)d_in[3];
    const int*   rb    = (const int*)d_in[4];
    float*       out   = (float*)d_out;
    const int n_points = in_sizes[2];

    // 1024 seg-blocks x 5 channel groups = 5120 waves; every output element
    // is written exactly once (empty cells get the zero accumulator), so no
    // pre-zero pass is needed.
    dim3 grid(BEV_CELLS / 16, C_CH / 16, 1);
    BevPoolV2_wmma_kernel<<<grid, dim3(32, 1, 1), 0, stream>>>(
        depth, feat, rd, rf, rb, out, n_points);
}